// _FPN_68427418960370
// MI455X (gfx1250) — compile-verified
//
#include <hip/hip_runtime.h>
#include <math.h>

#define POOLSZ 7
#define GRID 8
#define NGP 64            // 8x8 grid points (K dimension of pooling GEMM)
#define CCH 256           // channels
#define LROW 256          // floats per LDS grid-point row (exactly 64KB total)

typedef float v2f __attribute__((ext_vector_type(2)));
typedef float v8f __attribute__((ext_vector_type(8)));

// LDS index with per-row channel rotation: keeps the two half-waves of a WMMA
// A-fragment read (rows kb and kb+2) on disjoint 16-bank groups.
__device__ __forceinline__ int lds_idx(int g, int c) {
    return g * LROW + ((c + (g << 3)) & (LROW - 1));
}

__global__ __launch_bounds__(256)
void fpn_roialign_wmma(const float* __restrict__ p2,
                       const float* __restrict__ p3,
                       const float* __restrict__ p4,
                       const float* __restrict__ p5,
                       const float* __restrict__ rois,
                       const float* __restrict__ im_info,
                       float* __restrict__ out,
                       int R)
{
    extern __shared__ float vbuf[];   // [NGP][LROW] sampled grid, all channels

    const int r = blockIdx.x;
    if (r >= R) return;
    const int tid = threadIdx.x;

    // ---------------- per-RoI setup (wave-uniform) ----------------
    const float* roi = rois + (size_t)r * 5;
    const int   rb  = (int)roi[0];
    const float rx1 = roi[1], ry1 = roi[2], rx2 = roi[3], ry2 = roi[4];

    // FPN level: floor(log2(sqrt(h*w)/224) + 4), clipped to [2,5]
    const float hh = ry2 - ry1 + 1.0f;
    const float ww = rx2 - rx1 + 1.0f;
    float lf = floorf(log2f(sqrtf(hh * ww) * (1.0f / 224.0f)) + 4.0f);
    lf = fminf(fmaxf(lf, 2.0f), 5.0f);
    const int lvl = (int)lf;

    const float* fp; int H, W;
    switch (lvl) {
        case 2:  fp = p2; H = 256; W = 256; break;
        case 3:  fp = p3; H = 128; W = 128; break;
        case 4:  fp = p4; H = 64;  W = 64;  break;
        default: fp = p5; H = 32;  W = 32;  break;
    }
    const float scale = (float)H / im_info[0];

    const float x1 = rx1 * scale, y1 = ry1 * scale;
    const float x2 = rx2 * scale, y2 = ry2 * scale;

    int   y0i[GRID], y1i[GRID], x0i[GRID], x1i[GRID];
    float wy[GRID], wx[GRID];
    #pragma unroll
    for (int i = 0; i < GRID; ++i) {
        const float t  = (float)i * (1.0f / 7.0f);
        float ty = fminf(fmaxf(y1 + (y2 - y1) * t, 0.0f), (float)(H - 1));
        float tx = fminf(fmaxf(x1 + (x2 - x1) * t, 0.0f), (float)(W - 1));
        const float yf = floorf(ty), xf = floorf(tx);
        wy[i] = ty - yf;  wx[i] = tx - xf;
        y0i[i] = (int)yf; x0i[i] = (int)xf;
        y1i[i] = min(y0i[i] + 1, H - 1);
        x1i[i] = min(x0i[i] + 1, W - 1);
    }

    // -------- phase 1: bilinear-sample 8x8 grid for this thread's channel --------
    {
        const int c = tid;                                   // 256 threads == 256 channels
        const size_t base = ((size_t)rb * CCH + c) * (size_t)(H * W);
        #pragma unroll
        for (int gy = 0; gy < GRID; ++gy) {
            const size_t rowT = base + (size_t)y0i[gy] * W;
            const size_t rowB = base + (size_t)y1i[gy] * W;
            const float  fy   = wy[gy];
            #pragma unroll
            for (int gx = 0; gx < GRID; ++gx) {
                const float v00 = fp[rowT + x0i[gx]];
                const float v01 = fp[rowT + x1i[gx]];
                const float v10 = fp[rowB + x0i[gx]];
                const float v11 = fp[rowB + x1i[gx]];
                const float fx  = wx[gx];
                const float top = v00 + fx * (v01 - v00);
                const float bot = v10 + fx * (v11 - v10);
                vbuf[lds_idx(gy * GRID + gx, c)] = top + fy * (bot - top);
            }
        }
    }
    __syncthreads();

    // -------- phase 2: 2x2 avg-pool as GEMM  out[c,p] = sum_g v[c,g] * P[g,p] --------
    // via V_WMMA_F32_16X16X4_F32: M=16 channels, N=16 outputs (49 valid), K=4 per step.
    const int wave   = tid >> 5;
    const int lane   = tid & 31;
    const int half   = (lane < 16) ? 0 : 2;      // K offset for this half-wave
    const int col    = lane & 15;

    for (int m = wave; m < 16; m += 8) {         // two 16-channel M-tiles per wave
        const int ch = m * 16 + col;             // A-matrix row (channel) for this lane
        for (int n = 0; n < 4; ++n) {            // N-tiles (64 cols, 49 valid)
            const int  p     = n * 16 + col;
            const int  py    = p / 7;
            const int  px    = p - py * 7;
            const bool valid = (p < 49);
            v8f acc = {0.f, 0.f, 0.f, 0.f, 0.f, 0.f, 0.f, 0.f};
            #pragma unroll
            for (int j = 0; j < 16; ++j) {
                const int kb = 4 * j + half;     // this half-wave's K pair base
                v2f A, B;
                A[0] = vbuf[lds_idx(kb,     ch)];
                A[1] = vbuf[lds_idx(kb + 1, ch)];
                const int gy  = kb >> 3;
                const int gxe = kb & 7;          // even gx; A[1] is gx+1 (same gy row)
                const float wrow = (valid && (gy == py || gy == py + 1)) ? 0.25f : 0.0f;
                B[0] = ((gxe     == px) || (gxe     == px + 1)) ? wrow : 0.0f;
                B[1] = ((gxe + 1 == px) || (gxe + 1 == px + 1)) ? wrow : 0.0f;
                acc = __builtin_amdgcn_wmma_f32_16x16x4_f32(
                          /*neg_a=*/false, A, /*neg_b=*/false, B,
                          /*c_mod=*/(short)0, acc,
                          /*reuse_a=*/false, /*reuse_b=*/false);
            }
            if (valid) {
                // D layout: VGPR q holds rows M=q (lanes 0-15) / q+8 (lanes 16-31)
                const int chBase = m * 16 + ((lane < 16) ? 0 : 8);
                const size_t ob = ((size_t)r * CCH + chBase) * 49 + p;
                #pragma unroll
                for (int q = 0; q < 8; ++q)
                    out[ob + (size_t)q * 49] = acc[q];
            }
        }
    }
}

extern "C" void kernel_launch(void* const* d_in, const int* in_sizes, int n_in,
                              void* d_out, int out_size, void* d_ws, size_t ws_size,
                              hipStream_t stream) {
    const float* p2      = (const float*)d_in[0];
    const float* p3      = (const float*)d_in[1];
    const float* p4      = (const float*)d_in[2];
    const float* p5      = (const float*)d_in[3];
    const float* rois    = (const float*)d_in[4];
    const float* im_info = (const float*)d_in[5];
    float*       out     = (float*)d_out;

    const int R = in_sizes[4] / 5;                       // rois: [R,5]
    const size_t shmem = (size_t)NGP * LROW * sizeof(float);   // 64 KB

    fpn_roialign_wmma<<<R, 256, shmem, stream>>>(p2, p3, p4, p5, rois, im_info, out, R);
}